// TideTransformer_59811714564390
// MI455X (gfx1250) — compile-verified
//
#include <hip/hip_runtime.h>
#include <hip/hip_bf16.h>

typedef __attribute__((ext_vector_type(2))) float v2f;
typedef __attribute__((ext_vector_type(8))) float v8f;

// Problem constants (match reference)
#define MSDA_BS    8
#define MSDA_LQ    900
#define MSDA_NH    8
#define MSDA_LENV  21760
#define MSDA_ED    256

// ---------------------------------------------------------------------------
// GEMM: C[M,N] = A[M,256] @ W[256,N] + bias[N], f32 via V_WMMA_F32_16X16X4_F32
// Block = 256 threads (8 waves). Block tile: 64(M) x N. Wave w: m-strip w>>1,
// n-half w&1 (NTILES/2 n-tiles of 16). K staged in 32-wide LDS chunks.
//
// W is staged in a fragment-native interleaved layout: K-pair p stores
// (W[2p][n], W[2p+1][n]) adjacently at sW[p*SWS + 2n + {0,1}], so each WMMA
// B-fragment is a single aligned ds_load_b64 (pairs merge to ds_load_2addr).
// SWS % 64 == 32 makes the half-wave K-split (lanes 0-15 pair p, lanes 16-31
// pair p+1) land on complementary 32-bank sets -> conflict-free.
//
// Fragment loads are batched (af[]/bf[] arrays) so the scheduler can keep
// many ds_load_2addr_b64 in flight and amortize one s_wait_dscnt over 8
// WMMAs instead of stalling on LDS latency per WMMA pair.
// ---------------------------------------------------------------------------
template<int NTILES>
__global__ __launch_bounds__(256) void gemm_bias_wmma(
    const float* __restrict__ A, const float* __restrict__ W,
    const float* __restrict__ bias, float* __restrict__ C, int M)
{
    constexpr int N   = NTILES * 16;
    constexpr int K   = 256;
    constexpr int KC  = 32;          // K chunk in LDS
    constexpr int SWS = 2 * N + 32;  // dword stride per K-pair row (mod 64 == 32)
    constexpr int LDA = 34;          // pad: conflict-free A-fragment reads
    constexpr int BM  = 64;
    constexpr int WT  = NTILES / 2;  // n-tiles per wave

    __shared__ float sW[(KC / 2) * SWS];
    __shared__ float sA[BM * LDA];

    const int tid    = threadIdx.x;
    const int wave   = tid >> 5;
    const int lane   = tid & 31;
    const int half   = lane >> 4;    // K half-select per ISA fragment layout
    const int l16    = lane & 15;
    const int mstrip = wave >> 1;    // 0..3
    const int nhalf  = wave & 1;     // 0..1
    const int rowBase = blockIdx.x * BM;

    const v8f vzero = {0.f,0.f,0.f,0.f,0.f,0.f,0.f,0.f};
    v8f acc[WT];
#pragma unroll
    for (int t = 0; t < WT; ++t) acc[t] = vzero;

    for (int kc = 0; kc < K; kc += KC) {
        // Stage W[kc:kc+KC, 0:N] via float4 (B128) loads into interleaved layout
#pragma unroll
        for (int i = tid; i < (KC * N) / 4; i += 256) {
            const int e  = i * 4;
            const int kk = e / N;
            const int nn = e - kk * N;
            const float4 wv = *reinterpret_cast<const float4*>(W + (size_t)(kc + kk) * N + nn);
            float* d = &sW[(kk >> 1) * SWS + nn * 2 + (kk & 1)];
            d[0] = wv.x; d[2] = wv.y; d[4] = wv.z; d[6] = wv.w;
        }
        // Stage A[rowBase:rowBase+BM, kc:kc+KC] via float4 loads (rows clamped)
#pragma unroll
        for (int i = tid; i < (BM * KC) / 4; i += 256) {
            const int e  = i * 4;
            const int r  = e / KC;
            const int kk = e - r * KC;
            int gr = rowBase + r; if (gr > M - 1) gr = M - 1;
            const float4 av = *reinterpret_cast<const float4*>(A + (size_t)gr * K + kc + kk);
            float* d = &sA[r * LDA + kk];
            d[0] = av.x; d[1] = av.y; d[2] = av.z; d[3] = av.w;
        }
        // Prefetch next K-chunk (A is the 178MB roofline-critical stream)
        if (kc + KC < K) {
#pragma unroll
            for (int i = tid; i < (KC * N * 4) / 128; i += 256) {
                const int kk = (i * 32) / N;
                const int nn = (i * 32) % N;
                __builtin_prefetch(W + (size_t)(kc + KC + kk) * N + nn, 0, 0);
            }
            if (tid < BM) {
                int gr = rowBase + tid; if (gr > M - 1) gr = M - 1;
                __builtin_prefetch(A + (size_t)gr * K + kc + KC, 0, 0);
            }
        }
        __syncthreads();

        // Hoist all A fragments for this chunk (8 x b64, merge to 2addr pairs)
        v2f af[KC / 4];
#pragma unroll
        for (int k0 = 0; k0 < KC; k0 += 4) {
            af[k0 / 4] = *reinterpret_cast<const v2f*>(
                &sA[(mstrip * 16 + l16) * LDA + k0 + 2 * half]);
        }

#pragma unroll
        for (int k0 = 0; k0 < KC; k0 += 4) {
            const int pair = (k0 >> 1) + half;
            // Batch all B fragments of this k-step before any WMMA
            v2f bf[WT];
#pragma unroll
            for (int t = 0; t < WT; ++t) {
                const int n = (nhalf * WT + t) * 16 + l16;
                bf[t] = *reinterpret_cast<const v2f*>(&sW[pair * SWS + n * 2]);
            }
#pragma unroll
            for (int t = 0; t < WT; ++t) {
                acc[t] = __builtin_amdgcn_wmma_f32_16x16x4_f32(
                    false, af[k0 / 4], false, bf[t], (short)0, acc[t], false, false);
            }
        }
        __syncthreads();
    }

    // Epilogue: C/D layout — VGPR j: lanes 0-15 -> M=j, lanes 16-31 -> M=j+8
#pragma unroll
    for (int t = 0; t < WT; ++t) {
        const int n  = (nhalf * WT + t) * 16 + l16;
        const float bv = bias[n];
#pragma unroll
        for (int j = 0; j < 8; ++j) {
            const int m = rowBase + mstrip * 16 + half * 8 + j;
            if (m < M) C[(size_t)m * N + n] = acc[t][j] + bv;
        }
    }
}

// ---------------------------------------------------------------------------
// Softmax over trailing 16 (in place): rows = BS*LQ*NH
// ---------------------------------------------------------------------------
__global__ __launch_bounds__(256) void softmax16(float* __restrict__ aw, int rows)
{
    const int r = blockIdx.x * 256 + threadIdx.x;
    if (r >= rows) return;
    float* p = aw + (size_t)r * 16;
    float v[16];
    float mx = -3.402823466e38f;
#pragma unroll
    for (int i = 0; i < 16; ++i) { v[i] = p[i]; mx = fmaxf(mx, v[i]); }
    float s = 0.f;
#pragma unroll
    for (int i = 0; i < 16; ++i) { v[i] = expf(v[i] - mx); s += v[i]; }
    const float inv = 1.0f / s;
#pragma unroll
    for (int i = 0; i < 16; ++i) p[i] = v[i] * inv;
}

// ---------------------------------------------------------------------------
// Bilinear sampling + attention-weighted accumulation.
// One wave per (b,q,h) unit; lane = head-dim channel -> each corner gather is
// a contiguous 128B line (perfectly coalesced). Corner conditions are wave-
// uniform (scalar branch, no divergence).
//   v:   (BS, LEN_V, NH, 32)   ref: (BS, LQ, 4, 2)
//   off: (BS, LQ, 256=h*32+l*8+p*2+c)   aw: (BS*LQ*NH, 16=l*4+p)
//   out: (BS, LQ, NH, 32)
// ---------------------------------------------------------------------------
__global__ __launch_bounds__(256) void msda_sample(
    const float* __restrict__ v,  const float* __restrict__ ref,
    const float* __restrict__ off, const float* __restrict__ aw,
    float* __restrict__ out)
{
    const int LEVEL_HW[4]    = {128, 64, 32, 16};
    const int LEVEL_START[4] = {0, 16384, 20480, 21504};

    const int unit = blockIdx.x * 8 + (threadIdx.x >> 5);  // (b*LQ+q)*NH + h
    const int lane = threadIdx.x & 31;
    const int h  = unit & 7;
    const int bq = unit >> 3;
    const int b  = bq / MSDA_LQ;

    const float* offp = off + (size_t)bq * 256 + h * 32;
    const float* awp  = aw  + (size_t)unit * 16;
    const float* refp = ref + (size_t)bq * 8;
    const float* vb   = v   + (size_t)b * MSDA_LENV * 256;

    float acc = 0.f;
#pragma unroll
    for (int l = 0; l < 4; ++l) {
        const int H  = LEVEL_HW[l];
        const int Wd = LEVEL_HW[l];
        const float rx = refp[l * 2 + 0];
        const float ry = refp[l * 2 + 1];
        const float* vl = vb + (size_t)LEVEL_START[l] * 256;
#pragma unroll
        for (int p = 0; p < 4; ++p) {
            const float wa = awp[l * 4 + p];
            const float lx = rx + offp[l * 8 + p * 2 + 0] / (float)Wd;
            const float ly = ry + offp[l * 8 + p * 2 + 1] / (float)H;
            const float px = lx * (float)Wd - 0.5f;
            const float py = ly * (float)H  - 0.5f;
            const float fx = floorf(px), fy = floorf(py);
            const float dx = px - fx,    dy = py - fy;
            const int x0 = (int)fx, y0 = (int)fy;
            const float cw[4] = {(1.f-dx)*(1.f-dy), dx*(1.f-dy), (1.f-dx)*dy, dx*dy};
            const int   xs[4] = {x0, x0 + 1, x0,     x0 + 1};
            const int   ys[4] = {y0, y0,     y0 + 1, y0 + 1};
#pragma unroll
            for (int c = 0; c < 4; ++c) {
                const int xi = xs[c], yi = ys[c];
                if (xi >= 0 && xi < Wd && yi >= 0 && yi < H) {
                    const float* vp = vl + ((size_t)(yi * Wd + xi) * 8 + h) * 32;
                    acc = fmaf(wa * cw[c], vp[lane], acc);
                }
            }
        }
    }
    out[(size_t)unit * 32 + lane] = acc;
}

// ---------------------------------------------------------------------------
extern "C" void kernel_launch(void* const* d_in, const int* in_sizes, int n_in,
                              void* d_out, int out_size, void* d_ws, size_t ws_size,
                              hipStream_t stream)
{
    (void)in_sizes; (void)n_in; (void)out_size; (void)ws_size;
    const float* query   = (const float*)d_in[0];
    const float* refpts  = (const float*)d_in[1];
    const float* value   = (const float*)d_in[2];
    const float* w_value = (const float*)d_in[3];
    const float* b_value = (const float*)d_in[4];
    const float* w_off   = (const float*)d_in[5];
    const float* b_off   = (const float*)d_in[6];
    const float* w_attn  = (const float*)d_in[7];
    const float* b_attn  = (const float*)d_in[8];
    const float* w_out   = (const float*)d_in[9];
    const float* b_out   = (const float*)d_in[10];
    float* outp = (float*)d_out;

    constexpr int MV = MSDA_BS * MSDA_LENV;  // 174080
    constexpr int MQ = MSDA_BS * MSDA_LQ;    // 7200

    float* ws    = (float*)d_ws;
    float* vproj = ws;                             // MV*256 floats
    float* offb  = vproj + (size_t)MV * 256;       // MQ*256
    float* awb   = offb  + (size_t)MQ * 256;       // MQ*128
    float* sampb = awb   + (size_t)MQ * 128;       // MQ*256

    const dim3 blk(256);
    // 1) v = value @ w_value + b_value     (the 178MB stream; BW-critical)
    gemm_bias_wmma<16><<<dim3((MV + 63) / 64), blk, 0, stream>>>(value, w_value, b_value, vproj, MV);
    // 2) off = query @ w_off + b_off
    gemm_bias_wmma<16><<<dim3((MQ + 63) / 64), blk, 0, stream>>>(query, w_off, b_off, offb, MQ);
    // 3) aw_raw = query @ w_attn + b_attn
    gemm_bias_wmma<8><<<dim3((MQ + 63) / 64), blk, 0, stream>>>(query, w_attn, b_attn, awb, MQ);
    // 4) softmax over 16 (per b,q,h)
    softmax16<<<dim3((MQ * 8 + 255) / 256), blk, 0, stream>>>(awb, MQ * 8);
    // 5) bilinear sampling + weighted sum
    msda_sample<<<dim3(MQ), blk, 0, stream>>>(vproj, refpts, offb, awb, sampb);
    // 6) out = sampled @ w_out + b_out
    gemm_bias_wmma<16><<<dim3((MQ + 63) / 64), blk, 0, stream>>>(sampb, w_out, b_out, outp, MQ);
}